// GraphVAE_28570122453868
// MI455X (gfx1250) — compile-verified
//
#include <hip/hip_runtime.h>
#include <math.h>

#define NN 96
#define PITCH 100            // 100 words: 16B-aligned rows, stride%64=36 -> 16 distinct banks
#define MPM_ITERS 50
#define NT 1024              // 32 waves, one WGP

typedef __attribute__((ext_vector_type(2))) float v2f;
typedef __attribute__((ext_vector_type(8))) float v8f;

__global__ __launch_bounds__(NT, 1)
void graphvae_mpm_kernel(const float* __restrict__ A_gt,
                         const float* __restrict__ vec_logits,
                         float* __restrict__ out)
{
    __shared__ float sP[NN * PITCH];   // P' = A * dA_i * dA_j, zero diag
    __shared__ float sQ[NN * PITCH];   // Q  = B * dB_a * dB_b, zero diag (symmetric)
    __shared__ float sX[NN * PITCH];   // current X
    __shared__ float sM[NN * PITCH];   // M (phase1) then Xn (phase2b)
    __shared__ float sDegA[NN], sDegB[NN], sDA[NN], sDB[NN];
    __shared__ float sRed[33];

    const int tid  = threadIdx.x;
    const int lane = tid & 31;
    const int wave = tid >> 5;

    // ---------------- setup: stage A into sP, B=sigmoid(logits) into sQ ----------------
    for (int e = tid; e < NN * NN; e += NT) {
        int i = e / NN, j = e % NN;
        float l;
        if (i == j) {
            l = 1.0f;                                   // diag logits set to 1.0
        } else {
            int lo = (i < j) ? i : j;
            int hi = (i < j) ? j : i;
            int off = (lo * (2 * NN - lo - 1)) / 2 + (hi - lo - 1);  // triu_indices order
            l = vec_logits[off];
        }
        sQ[i * PITCH + j] = 1.0f / (1.0f + expf(-l));   // B
        sP[i * PITCH + j] = A_gt[e];                    // raw A for now
        sX[i * PITCH + j] = 1.0f / (float)NN;           // X0
    }
    __syncthreads();

    // degrees (over raw A, B) and diagonals
    if (tid < NN) {
        float ga = 0.f, gb = 0.f;
        for (int j = 0; j < NN; ++j) {
            ga += sP[tid * PITCH + j];
            gb += sQ[tid * PITCH + j];
        }
        sDA[tid] = sP[tid * PITCH + tid];
        sDB[tid] = sQ[tid * PITCH + tid];
        sDegA[tid] = ga;
        sDegB[tid] = gb;
    }
    __syncthreads();

    // scale: P' = A*dA_i*dA_j, Q = B*dB_a*dB_b, zero diagonals
    for (int e = tid; e < NN * NN; e += NT) {
        int i = e / NN, j = e % NN;
        float p = sP[i * PITCH + j] * sDA[i] * sDA[j];
        float q = sQ[i * PITCH + j] * sDB[i] * sDB[j];
        if (i == j) { p = 0.f; q = 0.f; }
        sP[i * PITCH + j] = p;
        sQ[i * PITCH + j] = q;
    }
    __syncthreads();

    // ---------------- MPM iterations (all LDS-resident) ----------------
    for (int it = 0; it < MPM_ITERS; ++it) {
        // phase 1: M[j,a] = max_b Q[a,b]*X[j,b]   (Q symmetric -> read Q[b,a], stride-1 lanes)
        for (int e = tid; e < NN * NN; e += NT) {
            int j = e / NN, a = e % NN;
            const float* xr = &sX[j * PITCH];
            float m = 0.0f;
            #pragma unroll 4
            for (int b = 0; b < NN; ++b)
                m = fmaxf(m, sQ[b * PITCH + a] * xr[b]);
            sM[j * PITCH + a] = m;
        }
        __syncthreads();

        // phase 2a: Y = P' @ M via V_WMMA_F32_16X16X4_F32 (accumulators in VGPRs)
        const int r = lane & 15;          // row-in-tile / col-in-tile
        const int h = lane >> 4;          // K-half selector
        v8f acc[2] = { {}, {} };
        const int nt = (wave + 32 < 36) ? 2 : 1;     // 36 tiles over 32 waves (wave-uniform)
        for (int q = 0; q < nt; ++q) {
            int t  = wave + 32 * q;
            int ti = t / 6, tn = t % 6;
            v8f c = {};
            for (int k0 = 0; k0 < NN; k0 += 4) {
                int ka = k0 + 2 * h;
                v2f av, bv;
                av.x = sP[(ti * 16 + r) * PITCH + ka];        // A 16x4: lane=row, VGPRs=K
                av.y = sP[(ti * 16 + r) * PITCH + ka + 1];
                bv.x = sM[ka * PITCH + tn * 16 + r];          // B 4x16: lane=col, VGPRs=K
                bv.y = sM[(ka + 1) * PITCH + tn * 16 + r];
                c = __builtin_amdgcn_wmma_f32_16x16x4_f32(
                        /*neg_a=*/false, av, /*neg_b=*/false, bv,
                        /*c_mod=*/(short)0, c, /*reuse_a=*/false, /*reuse_b=*/false);
            }
            acc[q] = c;
        }
        __syncthreads();   // everyone done reading sM -> safe to overwrite with Xn

        // phase 2b: Xn = X*NS + Y   (NS computed on the fly), stored into sM
        for (int q = 0; q < nt; ++q) {
            int t  = wave + 32 * q;
            int ti = t / 6, tn = t % 6;
            int col = tn * 16 + r;
            #pragma unroll
            for (int v = 0; v < 8; ++v) {
                int row = ti * 16 + v + 8 * h;                 // C/D layout: VGPR v, half h
                float ns = sDA[row] * sDB[col] /
                           (fabsf(sDegA[row] - sDegB[col]) + 1.0f);
                sM[row * PITCH + col] = sX[row * PITCH + col] * ns + acc[q][v];
            }
        }
        __syncthreads();

        // phase 3: Frobenius normalize
        float ss = 0.f;
        for (int e = tid; e < NN * NN; e += NT) {
            float v = sM[(e / NN) * PITCH + (e % NN)];
            ss += v * v;
        }
        #pragma unroll
        for (int msk = 16; msk >= 1; msk >>= 1)
            ss += __shfl_xor(ss, msk, 32);
        if (lane == 0) sRed[wave] = ss;
        __syncthreads();
        if (tid == 0) {
            float tot = 0.f;
            for (int w = 0; w < 32; ++w) tot += sRed[w];
            sRed[32] = 1.0f / sqrtf(tot);
        }
        __syncthreads();
        float inv = sRed[32];
        for (int e = tid; e < NN * NN; e += NT)
            sX[(e / NN) * PITCH + (e % NN)] =
                sM[(e / NN) * PITCH + (e % NN)] * inv;
        __syncthreads();
    }

    // ---------------- writeback ----------------
    for (int e = tid; e < NN * NN; e += NT)
        out[e] = sX[(e / NN) * PITCH + (e % NN)];
}

extern "C" void kernel_launch(void* const* d_in, const int* in_sizes, int n_in,
                              void* d_out, int out_size, void* d_ws, size_t ws_size,
                              hipStream_t stream) {
    (void)in_sizes; (void)n_in; (void)d_ws; (void)ws_size; (void)out_size;
    const float* A_gt = (const float*)d_in[0];        // 96*96 f32
    const float* vec  = (const float*)d_in[1];        // 96*95/2 f32
    float* out = (float*)d_out;                        // 96*96 f32
    graphvae_mpm_kernel<<<1, NT, 0, stream>>>(A_gt, vec, out);
}